// GELU259_23648089932077
// MI455X (gfx1250) — compile-verified
//
#include <hip/hip_runtime.h>
#include <hip/hip_bf16.h>
#include <math.h>

// ---------------------------------------------------------------------------
// Fused "GELU + prototype-injection" pipeline for MI455X (gfx1250, wave32).
// Bandwidth-bound: ~192 MB of HBM traffic -> ~8.3 us at 23.3 TB/s.
// The buf @ y_mean similarity matvec runs on V_WMMA_F32_16X16X4_F32.
// ---------------------------------------------------------------------------

typedef float v2f __attribute__((ext_vector_type(2)));
typedef float v8f __attribute__((ext_vector_type(8)));

#define FIRE_THRESH 0.85f
#define NUDGE_RATE  2.0f
#define SQRT_2_OVER_PI 0.7978845608028654f
#define EPS_NORM 1e-12f
#define K1_ROWS 64

__device__ __forceinline__ float gelu_f(float v) {
    float inner = SQRT_2_OVER_PI * (v + 0.044715f * v * v * v);
    return 0.5f * v * (1.0f + tanhf(inner));
}

// ---------------------------------------------------------------------------
// Kernel 1: column partial sums of gelu(x). Each block: 64 rows x D cols.
// Thread t owns contiguous cols [8t, 8t+8) -> float4 coalesced loads.
// ---------------------------------------------------------------------------
__global__ void k1_colsum(const float* __restrict__ x, float* __restrict__ partials,
                          int D) {
    int t  = threadIdx.x;
    int c0 = t * 8;
    if (c0 + 8 > D) return;
    size_t r0 = (size_t)blockIdx.x * K1_ROWS;

    float acc[8];
#pragma unroll
    for (int j = 0; j < 8; ++j) acc[j] = 0.0f;

    for (int r = 0; r < K1_ROWS; ++r) {
        const float* row = x + (r0 + r) * (size_t)D + c0;
        __builtin_prefetch(row + D, 0, 1);           // global_prefetch_b8 next row
        float4 p0 = *(const float4*)(row);
        float4 p1 = *(const float4*)(row + 4);
        acc[0] += gelu_f(p0.x); acc[1] += gelu_f(p0.y);
        acc[2] += gelu_f(p0.z); acc[3] += gelu_f(p0.w);
        acc[4] += gelu_f(p1.x); acc[5] += gelu_f(p1.y);
        acc[6] += gelu_f(p1.z); acc[7] += gelu_f(p1.w);
    }
    float* op = partials + (size_t)blockIdx.x * D + c0;
    *(float4*)(op)     = make_float4(acc[0], acc[1], acc[2], acc[3]);
    *(float4*)(op + 4) = make_float4(acc[4], acc[5], acc[6], acc[7]);
}

// ---------------------------------------------------------------------------
// Kernel 2: reduce partials -> y_mean [D], and ||y_mean|| (single block).
// ---------------------------------------------------------------------------
__global__ void k2_reduce(const float* __restrict__ partials, int nblk, int D,
                          float invBT, float* __restrict__ ymean,
                          float* __restrict__ mnorm) {
    __shared__ float red[256];
    int t = threadIdx.x;
    float sq = 0.0f;
    for (int c = t; c < D; c += 256) {
        float s = 0.0f;
        for (int b = 0; b < nblk; ++b) s += partials[(size_t)b * D + c];
        float m = s * invBT;
        ymean[c] = m;
        sq += m * m;
    }
    red[t] = sq; __syncthreads();
    for (int off = 128; off > 0; off >>= 1) {
        if (t < off) red[t] += red[t + off];
        __syncthreads();
    }
    if (t == 0) mnorm[0] = sqrtf(red[0]);
}

// ---------------------------------------------------------------------------
// Kernel 3: sims[n] = mask ? dot(buf[n], y_mean)/(max(||buf[n]||,eps)*max(||m||,eps)) : -1
// One wave (32 threads) per 16-row group. Dot products via WMMA f32 16x16x4:
//   A = 16x4 tile of buf (lanes 0-15: K0/K1 in v[0]/v[1]; lanes 16-31: K2/K3)
//   B = y_mean chunk replicated across all 16 columns -> every D column = dot.
// C/D layout: lane 0 holds rows M=0..7 in c[0..7], lane 16 holds M=8..15.
// ---------------------------------------------------------------------------
__global__ void k3_sims(const float* __restrict__ buf,
                        const unsigned char* __restrict__ mask,
                        const float* __restrict__ ymean,
                        const float* __restrict__ mnorm_p,
                        float* __restrict__ sims, int D) {
    int lane = threadIdx.x;           // 0..31
    int r0   = blockIdx.x * 16;
    int half = lane >> 4;             // 0: K0/K1, 1: K2/K3
    int mrow = lane & 15;
    const float* arow = buf + ((size_t)(r0 + mrow)) * D + half * 2;
    const float* mvec = ymean + half * 2;

    v8f c = {0.f, 0.f, 0.f, 0.f, 0.f, 0.f, 0.f, 0.f};
    for (int k0 = 0; k0 < D; k0 += 4) {
        v2f a, b;
        a.x = arow[k0 + 0];
        a.y = arow[k0 + 1];
        b.x = mvec[k0 + 0];
        b.y = mvec[k0 + 1];
        c = __builtin_amdgcn_wmma_f32_16x16x4_f32(
                /*neg_a=*/false, a, /*neg_b=*/false, b,
                /*c_mod=*/(short)0, c, /*reuse_a=*/false, /*reuse_b=*/false);
    }

    __shared__ float dots[16];
    __shared__ float norms[16];
    if (lane == 0) {
#pragma unroll
        for (int v = 0; v < 8; ++v) dots[v] = c[v];
    }
    if (lane == 16) {
#pragma unroll
        for (int v = 0; v < 8; ++v) dots[8 + v] = c[v];
    }
    // Row L2 norms: all 32 lanes cooperate per row, butterfly reduce (wave32).
    for (int r = 0; r < 16; ++r) {
        const float* row = buf + ((size_t)(r0 + r)) * D;
        float s = 0.0f;
        for (int k = lane; k < D; k += 32) { float v = row[k]; s += v * v; }
        for (int off = 16; off > 0; off >>= 1) s += __shfl_xor(s, off, 32);
        if (lane == 0) norms[r] = sqrtf(s);
    }
    __syncthreads();
    float mn = fmaxf(mnorm_p[0], EPS_NORM);
    if (lane < 16) {
        int r = r0 + lane;
        float bn = fmaxf(norms[lane], EPS_NORM);
        sims[r] = mask[r] ? (dots[lane] / (bn * mn)) : -1.0f;
    }
}

// ---------------------------------------------------------------------------
// Kernel 4 (single block): argmax(sims) w/ first-index tie-break, injection
// level logic, nudge_unit[D] and sn_n[D] (normalized stored prototype).
// ---------------------------------------------------------------------------
__global__ void k4_scalar(const float* __restrict__ sims,
                          const float* __restrict__ inj,
                          const float* __restrict__ log_inject,
                          const float* __restrict__ xbuf,
                          const float* __restrict__ xglobal,
                          int N, int D,
                          float* __restrict__ nudge_unit,
                          float* __restrict__ sn_n,
                          float* __restrict__ scal) {
    __shared__ float rv[256];
    __shared__ float r2[256];
    __shared__ int   ri[256];
    __shared__ int   s_idx;
    __shared__ float s_ndn, s_sxn;

    int t = threadIdx.x;
    float bv = -1e30f; int bi = 0;
    for (int j = t; j < N; j += 256) {
        float v = sims[j];
        if (v > bv) { bv = v; bi = j; }   // increasing j keeps first max
    }
    rv[t] = bv; ri[t] = bi; __syncthreads();
    for (int off = 128; off > 0; off >>= 1) {
        if (t < off) {
            float v = rv[t + off]; int i = ri[t + off];
            if (v > rv[t] || (v == rv[t] && i < ri[t])) { rv[t] = v; ri[t] = i; }
        }
        __syncthreads();
    }
    if (t == 0) {
        int idx = ri[0];
        float max_sim = rv[0];
        float inject_base = fminf(fmaxf(expf(log_inject[0]), 0.001f), 2.0f);
        float inj_near = inj[idx];
        bool fired = max_sim > FIRE_THRESH;
        float lvl = fired ? ((inj_near < 1e-6f) ? inject_base : inj_near * NUDGE_RATE)
                          : inj_near;
        s_idx = idx;
        scal[0] = lvl;
    }
    __syncthreads();

    const float* sx = xbuf + (size_t)s_idx * D;
    float a_nd = 0.0f, a_sx = 0.0f;
    for (int c = t; c < D; c += 256) {
        float s  = sx[c];
        float nd = s - xglobal[c];
        a_nd += nd * nd;
        a_sx += s * s;
    }
    rv[t] = a_nd; r2[t] = a_sx; __syncthreads();
    for (int off = 128; off > 0; off >>= 1) {
        if (t < off) { rv[t] += rv[t + off]; r2[t] += r2[t + off]; }
        __syncthreads();
    }
    if (t == 0) {
        s_ndn = fmaxf(sqrtf(rv[0]), 1e-6f);
        s_sxn = fmaxf(sqrtf(r2[0]), EPS_NORM);
    }
    __syncthreads();
    float ndn = s_ndn, sxn = s_sxn;
    for (int c = t; c < D; c += 256) {
        float s = sx[c];
        nudge_unit[c] = (s - xglobal[c]) / ndn;
        sn_n[c]       = s / sxn;
    }
}

// ---------------------------------------------------------------------------
// Kernel 5: one block per token. tok_sim = clip(dot(x,sn_n)/max(||x||,eps),0,1),
// out = gelu(x + scale*nudge_unit), scale = (lvl<1e-6) ? 0 : lvl*tok_sim.
// scale==0 reproduces y_plain = gelu(x) bit-exactly.
// ---------------------------------------------------------------------------
__global__ void k5_out(const float* __restrict__ x,
                       const float* __restrict__ nudge_unit,
                       const float* __restrict__ sn_n,
                       const float* __restrict__ scal,
                       float* __restrict__ out, int D) {
    __shared__ float rd[256];
    __shared__ float rs[256];
    __shared__ float s_scale;

    int t  = threadIdx.x;
    int c0 = t * 8;
    if (c0 + 8 > D) return;
    size_t row = blockIdx.x;
    const float* xr = x + row * (size_t)D;
    float*       yr = out + row * (size_t)D;

    float4 p0 = *(const float4*)(xr + c0);
    float4 p1 = *(const float4*)(xr + c0 + 4);
    float4 s0 = *(const float4*)(sn_n + c0);
    float4 s1 = *(const float4*)(sn_n + c0 + 4);
    float xv[8] = {p0.x, p0.y, p0.z, p0.w, p1.x, p1.y, p1.z, p1.w};
    float sv[8] = {s0.x, s0.y, s0.z, s0.w, s1.x, s1.y, s1.z, s1.w};

    float dt = 0.0f, sq = 0.0f;
#pragma unroll
    for (int j = 0; j < 8; ++j) { dt += xv[j] * sv[j]; sq += xv[j] * xv[j]; }
    rd[t] = dt; rs[t] = sq; __syncthreads();
    for (int off = 128; off > 0; off >>= 1) {
        if (t < off) { rd[t] += rd[t + off]; rs[t] += rs[t + off]; }
        __syncthreads();
    }
    if (t == 0) {
        float lvl = scal[0];
        float xn  = fmaxf(sqrtf(rs[0]), EPS_NORM);
        float ts  = fminf(fmaxf(rd[0] / xn, 0.0f), 1.0f);
        s_scale = (lvl < 1e-6f) ? 0.0f : lvl * ts;
    }
    __syncthreads();
    float sc = s_scale;

    float4 n0 = *(const float4*)(nudge_unit + c0);
    float4 n1 = *(const float4*)(nudge_unit + c0 + 4);
    float nv[8] = {n0.x, n0.y, n0.z, n0.w, n1.x, n1.y, n1.z, n1.w};
    float yv[8];
#pragma unroll
    for (int j = 0; j < 8; ++j) yv[j] = gelu_f(xv[j] + sc * nv[j]);
    *(float4*)(yr + c0)     = make_float4(yv[0], yv[1], yv[2], yv[3]);
    *(float4*)(yr + c0 + 4) = make_float4(yv[4], yv[5], yv[6], yv[7]);
}

// ---------------------------------------------------------------------------
extern "C" void kernel_launch(void* const* d_in, const int* in_sizes, int n_in,
                              void* d_out, int out_size, void* d_ws, size_t ws_size,
                              hipStream_t stream) {
    const float*         x          = (const float*)d_in[0];
    const float*         log_inject = (const float*)d_in[1];
    const float*         buf        = (const float*)d_in[2];
    const float*         xbuf       = (const float*)d_in[3];
    const float*         inj        = (const float*)d_in[4];
    const float*         xglobal    = (const float*)d_in[5];
    const unsigned char* mask       = (const unsigned char*)d_in[6];
    float*               out        = (float*)d_out;

    const int D  = in_sizes[5];            // 2048
    const int BT = in_sizes[0] / D;        // 8192
    const int N  = in_sizes[4];            // 512
    const int nblk1 = BT / K1_ROWS;        // 128

    // Workspace layout (floats), all written-before-read, no atomics.
    float* ws       = (float*)d_ws;
    float* partials = ws;                                  // nblk1 * D
    float* ymean    = partials + (size_t)nblk1 * D;        // D
    float* mnorm    = ymean + D;                           // 1
    float* sims     = mnorm + 1;                           // N
    float* scal     = sims + N;                            // 4
    float* nudge    = scal + 4;                            // D
    float* snn      = nudge + D;                           // D

    k1_colsum<<<nblk1, 256, 0, stream>>>(x, partials, D);
    k2_reduce<<<1, 256, 0, stream>>>(partials, nblk1, D, 1.0f / (float)BT,
                                     ymean, mnorm);
    k3_sims<<<N / 16, 32, 0, stream>>>(buf, mask, ymean, mnorm, sims, D);
    k4_scalar<<<1, 256, 0, stream>>>(sims, inj, log_inject, xbuf, xglobal,
                                     N, D, nudge, snn, scal);
    k5_out<<<BT, 256, 0, stream>>>(x, nudge, snn, scal, out, D);
}